// Bilevel_35347580846320
// MI455X (gfx1250) — compile-verified
//
#include <hip/hip_runtime.h>
#include <hip/hip_bf16.h>
#include <math.h>

// GroupProx: val = c.x + MU * sum_g sqrt(segsum(x^2) + EPS), P=2,Q=1 -> q/p = 0.5
#define MU_C  0.1f
#define EPS_C 1e-8f

typedef __attribute__((ext_vector_type(2))) float v2f;
typedef __attribute__((ext_vector_type(8))) float v8f;

// Full-wave (32-lane) sum using V_WMMA_F32_16X16X4_F32 with a ones B-matrix.
// A layout (16x4 f32): lanes 0-15 hold {K0,K1} for M=lane, lanes 16-31 hold {K2,K3}.
// With a = {v, 0}: rowsum(m) = v(lane m) + v(lane m+16). D VGPR r on lanes 0-15 is
// D[r][n] = rowsum(r); on lanes 16-31 it is rowsum(8+r). Summing the 8 D registers
// gives half-sums; shfl_xor(16) completes the wave sum (result in all lanes).
__device__ __forceinline__ float wave_sum_wmma(float v) {
  v2f a; a[0] = v;    a[1] = 0.0f;
  v2f b; b[0] = 1.0f; b[1] = 1.0f;
  v8f cz = {};
  v8f d = __builtin_amdgcn_wmma_f32_16x16x4_f32(
      /*neg_a=*/false, a, /*neg_b=*/false, b,
      /*c_mod=*/(short)0, cz, /*reuse_a=*/false, /*reuse_b=*/false);
  float s = d[0] + d[1] + d[2] + d[3] + d[4] + d[5] + d[6] + d[7];
  s += __shfl_xor(s, 16, 32);
  return s;
}

__global__ void zero_ws_kernel(float* __restrict__ ws, const int* __restrict__ ngp) {
  const int ntot = *ngp + 1;  // pow sums + dot slot
  for (int i = blockIdx.x * blockDim.x + threadIdx.x; i < ntot;
       i += gridDim.x * blockDim.x)
    ws[i] = 0.0f;
}

// Each wave processes contiguous 128-element tiles. Per lane: 4 sorted keys ->
// head run / interior / tail run; wave-level head-flag segmented inclusive scan
// chains tail runs across lanes so each maximal chain costs one atomicAdd.
__global__ void groupprox_main(const float* __restrict__ x, const float* __restrict__ c,
                               const int* __restrict__ seg, const int* __restrict__ ngp,
                               float* __restrict__ ws, int n) {
  const int nGroups = *ngp;
  float* __restrict__ pw = ws;
  float* __restrict__ dw = ws + nGroups;

  const unsigned lane = threadIdx.x & 31u;
  const int wave   = (int)((blockIdx.x * blockDim.x + threadIdx.x) >> 5);
  const int nWaves = (int)((gridDim.x * blockDim.x) >> 5);
  const int nTiles = (n + 127) >> 7;

  float dotAcc = 0.0f;

  for (int t = wave; t < nTiles; t += nWaves) {
    const int base = (t << 7) + ((int)lane << 2);

    // Prefetch next tile for this wave (gfx1250 global_prefetch_b8).
    const int tn = t + nWaves;
    if (tn < nTiles) {
      const int pb = (tn << 7) + ((int)lane << 2);
      __builtin_prefetch(x + pb, 0, 0);
      __builtin_prefetch(c + pb, 0, 0);
      __builtin_prefetch(seg + pb, 0, 0);
    }

    float x0, x1, x2, x3, c0, c1, c2, c3;
    int s0, s1, s2, s3;
    if (base + 3 < n) {                       // fast path: aligned b128 loads
      const float4 xv = *reinterpret_cast<const float4*>(x + base);
      const float4 cv = *reinterpret_cast<const float4*>(c + base);
      const int4   sv = *reinterpret_cast<const int4*>(seg + base);
      x0 = xv.x; x1 = xv.y; x2 = xv.z; x3 = xv.w;
      c0 = cv.x; c1 = cv.y; c2 = cv.z; c3 = cv.w;
      s0 = sv.x; s1 = sv.y; s2 = sv.z; s3 = sv.w;
    } else {                                  // guarded tail
      const int last = n - 1;
      const int i0 = base, i1 = base + 1, i2 = base + 2, i3 = base + 3;
      x0 = (i0 < n) ? x[i0] : 0.0f; c0 = (i0 < n) ? c[i0] : 0.0f;
      x1 = (i1 < n) ? x[i1] : 0.0f; c1 = (i1 < n) ? c[i1] : 0.0f;
      x2 = (i2 < n) ? x[i2] : 0.0f; c2 = (i2 < n) ? c[i2] : 0.0f;
      x3 = (i3 < n) ? x[i3] : 0.0f; c3 = (i3 < n) ? c[i3] : 0.0f;
      s0 = seg[(i0 < n) ? i0 : last];
      s1 = seg[(i1 < n) ? i1 : last];
      s2 = seg[(i2 < n) ? i2 : last];
      s3 = seg[(i3 < n) ? i3 : last];
    }

    dotAcc = fmaf(x0, c0, fmaf(x1, c1, fmaf(x2, c2, fmaf(x3, c3, dotAcc))));

    const float q0 = x0 * x0, q1 = x1 * x1, q2 = x2 * x2, q3 = x3 * x3;

    // Trailing run (keys are sorted non-decreasing).
    float tailSum = q3;
    if (s2 == s3) { tailSum += q2; if (s1 == s3) { tailSum += q1; if (s0 == s3) tailSum += q0; } }
    const bool pure = (s0 == s3);

    // Segmented inclusive scan of tail runs across the wave.
    const int prevTailKey = __shfl_up(s3, 1, 32);
    const int connect = (pure && lane != 0u && prevTailKey == s0) ? 1 : 0;
    float v = tailSum;
    int flag = 1 - connect;
#pragma unroll
    for (int dlt = 1; dlt < 32; dlt <<= 1) {
      const float vp = __shfl_up(v, dlt, 32);
      const int   fp = __shfl_up(flag, dlt, 32);
      if ((int)lane >= dlt) {
        if (!flag) v += vp;
        flag |= fp;
      }
    }
    const float prevScan = __shfl_up(v, 1, 32);

    // Head-run flush (chains from previous lanes fold in here) + interior runs.
    if (!pure) {
      float h = q0;
      if (s1 == s0) { h += q1; if (s2 == s0) h += q2; }
      if (lane != 0u && prevTailKey == s0) h += prevScan;
      atomicAdd(&pw[s0], h);
      if (s1 != s0 && s1 != s3) atomicAdd(&pw[s1], q1);
      if (s2 != s0 && s2 != s3) atomicAdd(&pw[s2], q2);
    }

    // Chain termination: flush scanned tail if the next lane does not continue it.
    const int nextHeadKey = __shfl_down(s0, 1, 32);
    const bool term = (lane == 31u) || (nextHeadKey != s3);
    if (term) atomicAdd(&pw[s3], v);
  }

  // Dot-product reduction via WMMA (all lanes active; EXEC is full here).
  const float dTot = wave_sum_wmma(dotAcc);
  if (lane == 0u) atomicAdd(dw, dTot);
}

__global__ void groupprox_final(const float* __restrict__ ws, const int* __restrict__ ngp,
                                float* __restrict__ out) {
  const int nGroups = *ngp;
  const float dotv = ws[nGroups];

  float acc = 0.0f;
  for (int g = (int)threadIdx.x; g < nGroups; g += (int)blockDim.x)
    acc += sqrtf(ws[g] + EPS_C);   // (s + eps)^(q/p) with q/p = 0.5

  const float wsum = wave_sum_wmma(acc);

  __shared__ float red[32];
  const unsigned lane = threadIdx.x & 31u;
  const unsigned w    = threadIdx.x >> 5;
  if (lane == 0u) red[w] = wsum;
  __syncthreads();
  if (threadIdx.x == 0u) {
    float tot = 0.0f;
    const unsigned nw = blockDim.x >> 5;
    for (unsigned i = 0; i < nw; ++i) tot += red[i];
    out[0] = dotv + MU_C * tot;
  }
}

extern "C" void kernel_launch(void* const* d_in, const int* in_sizes, int n_in,
                              void* d_out, int out_size, void* d_ws, size_t ws_size,
                              hipStream_t stream) {
  const float* x   = (const float*)d_in[0];
  const float* c   = (const float*)d_in[1];
  const int*   seg = (const int*)d_in[2];
  const int*   ngp = (const int*)d_in[3];   // n_groups lives on device (scalar input)
  float* ws  = (float*)d_ws;
  float* out = (float*)d_out;
  const int n = in_sizes[0];

  zero_ws_kernel<<<256, 256, 0, stream>>>(ws, ngp);
  groupprox_main<<<1024, 256, 0, stream>>>(x, c, seg, ngp, ws, n);
  groupprox_final<<<1, 256, 0, stream>>>(ws, ngp, out);
}